// MultiClassHingeLoss_9268539425374
// MI455X (gfx1250) — compile-verified
//
#include <hip/hip_runtime.h>

// ---------------------------------------------------------------------------
// Multi-class hinge loss, MI455X (gfx1250, wave32).
//   pred  : [B, C] f32   (B=65536, C=1000)
//   target: [B]    i64
//   out   : scalar f32 = mean_b max(0, max_{j!=y} (pred[b,j] - pred[b,y] + 1))
//
// Bandwidth-bound: 262 MB read @ 23.3 TB/s => ~11 us floor. Kernel 1 streams
// pred with coalesced non-temporal float4 loads (one wave32 per 4000-B row) and
// writes 8192 deterministic per-block partial sums. Kernel 2 reduces the
// partials through the matrix pipe with V_WMMA_F32_16X16X4_F32 (D = A*1 + C),
// 64 f32 summed per instruction, fixed order => deterministic. Bulk chunks are
// loaded unpredicated (global_load_b64) so 4 loads stay in flight per 4-WMMA
// group; only a possible partial tail chunk is predicated.
// ---------------------------------------------------------------------------

typedef float vfloat4 __attribute__((ext_vector_type(4)));
typedef float v2f     __attribute__((ext_vector_type(2)));
typedef float v8f     __attribute__((ext_vector_type(8)));

#define NEG_INF (-__builtin_huge_valf())

// ---------------------------------------------------------------------------
// Kernel 1: one wave32 per row. NVEC = C/4 (compile-time when 250 -> full
// unroll, 8 float4 loads issued ahead of the dependent max chain).
// ---------------------------------------------------------------------------
template <int NVEC>
__global__ __launch_bounds__(256)
void hinge_rows_kernel(const float* __restrict__ pred,
                       const long long* __restrict__ target,
                       float* __restrict__ partial,
                       int B, int C) {
    __shared__ float swave[8];
    const int lane = threadIdx.x & 31;
    const int wave = threadIdx.x >> 5;
    const int row  = blockIdx.x * 8 + wave;

    float per_sample = 0.0f;
    if (row < B) {
        const int    y       = (int)target[row];
        const float* rowp    = pred + (size_t)row * (size_t)C;
        const float  correct = rowp[y];           // uniform per wave, one L2 req
        float acc = NEG_INF;

        if (NVEC > 0) {
            const vfloat4* rowv = (const vfloat4*)rowp;
#pragma unroll
            for (int it = 0; it < (NVEC + 31) / 32; ++it) {
                const int idx = it * 32 + lane;
                if (idx < NVEC) {
                    vfloat4 v = __builtin_nontemporal_load(&rowv[idx]); // TH_NT: no reuse
                    const int col = idx << 2;
                    acc = fmaxf(acc, (col     == y) ? NEG_INF : v[0]);
                    acc = fmaxf(acc, (col + 1 == y) ? NEG_INF : v[1]);
                    acc = fmaxf(acc, (col + 2 == y) ? NEG_INF : v[2]);
                    acc = fmaxf(acc, (col + 3 == y) ? NEG_INF : v[3]);
                }
            }
        } else {
            const int nvec = C >> 2;
            const vfloat4* rowv = (const vfloat4*)rowp;
            for (int idx = lane; idx < nvec; idx += 32) {
                vfloat4 v = __builtin_nontemporal_load(&rowv[idx]);
                const int col = idx << 2;
                acc = fmaxf(acc, (col     == y) ? NEG_INF : v[0]);
                acc = fmaxf(acc, (col + 1 == y) ? NEG_INF : v[1]);
                acc = fmaxf(acc, (col + 2 == y) ? NEG_INF : v[2]);
                acc = fmaxf(acc, (col + 3 == y) ? NEG_INF : v[3]);
            }
            for (int j = (nvec << 2) + lane; j < C; j += 32) {
                const float v = rowp[j];
                acc = fmaxf(acc, (j == y) ? NEG_INF : v);
            }
        }

        // wave32 butterfly max-reduce
#pragma unroll
        for (int off = 16; off > 0; off >>= 1)
            acc = fmaxf(acc, __shfl_xor(acc, off, 32));

        per_sample = fmaxf(acc - correct + 1.0f, 0.0f);
    }

    if (lane == 0) swave[wave] = per_sample;
    __syncthreads();
    if (threadIdx.x == 0) {
        float s = 0.0f;
#pragma unroll
        for (int i = 0; i < 8; ++i) s += swave[i];
        partial[blockIdx.x] = s;   // fixed order per block => deterministic
    }
}

// ---------------------------------------------------------------------------
// Kernel 2: single wave32. Sum n partials via V_WMMA_F32_16X16X4_F32 with an
// all-ones B: D[m,n] = sum_k A[m,k]. Each WMMA folds a 64-float chunk; the
// (M,K) placement of values is irrelevant since everything is summed. Four
// independent accumulators cover XDL latency. Deterministic fixed order.
// ---------------------------------------------------------------------------
__global__ __launch_bounds__(32)
void reduce_mean_wmma(const float* __restrict__ partial,
                      float* __restrict__ out,
                      int n, float invB) {
    const int lane  = threadIdx.x;          // 32 threads, EXEC all-ones
    const int nfull = n >> 6;               // complete 64-float chunks

#if __has_builtin(__builtin_amdgcn_wmma_f32_16x16x4_f32)
    v2f ones; ones[0] = 1.0f; ones[1] = 1.0f;
    v8f acc0 = {}, acc1 = {}, acc2 = {}, acc3 = {};

    const v2f* __restrict__ p2 = (const v2f*)partial;  // lane owns 8B / chunk

    int c = 0;
    // Bulk: unpredicated b64 loads, 4 chunks in flight per 4-WMMA group.
    for (; c + 4 <= nfull; c += 4) {
        v2f a0 = p2[((c + 0) << 5) + lane];
        v2f a1 = p2[((c + 1) << 5) + lane];
        v2f a2 = p2[((c + 2) << 5) + lane];
        v2f a3 = p2[((c + 3) << 5) + lane];
        acc0 = __builtin_amdgcn_wmma_f32_16x16x4_f32(false, a0, false, ones, (short)0, acc0, false, false);
        acc1 = __builtin_amdgcn_wmma_f32_16x16x4_f32(false, a1, false, ones, (short)0, acc1, false, false);
        acc2 = __builtin_amdgcn_wmma_f32_16x16x4_f32(false, a2, false, ones, (short)0, acc2, false, false);
        acc3 = __builtin_amdgcn_wmma_f32_16x16x4_f32(false, a3, false, ones, (short)0, acc3, false, false);
    }
    for (; c < nfull; ++c) {
        v2f a = p2[(c << 5) + lane];
        acc0 = __builtin_amdgcn_wmma_f32_16x16x4_f32(false, a, false, ones, (short)0, acc0, false, false);
    }
    // Tail: at most one partial chunk, predicated loads (uniform outer branch,
    // EXEC is all-ones again by WMMA time).
    if (n & 63) {
        const int e = (nfull << 6) + (lane << 1);
        v2f a;
        a[0] = (e     < n) ? partial[e]     : 0.0f;
        a[1] = (e + 1 < n) ? partial[e + 1] : 0.0f;
        acc0 = __builtin_amdgcn_wmma_f32_16x16x4_f32(false, a, false, ones, (short)0, acc0, false, false);
    }

    v8f d = (acc0 + acc1) + (acc2 + acc3);
    // Lane layout of D (16x16 f32): lanes 0-15 hold row sums M=0..7 in d[0..7],
    // lanes 16-31 hold M=8..15; every column n is identical.
    float s = d[0] + d[1] + d[2] + d[3] + d[4] + d[5] + d[6] + d[7];
    const float total = __shfl(s, 0, 32) + __shfl(s, 16, 32);
#else
    // Fallback: deterministic strided + butterfly sum (still fixed order).
    float s = 0.0f;
    for (int i = lane; i < n; i += 32) s += partial[i];
#pragma unroll
    for (int off = 16; off > 0; off >>= 1) s += __shfl_xor(s, off, 32);
    const float total = s;
#endif

    if (lane == 0) out[0] = total * invB;
}

// ---------------------------------------------------------------------------
extern "C" void kernel_launch(void* const* d_in, const int* in_sizes, int n_in,
                              void* d_out, int out_size, void* d_ws, size_t ws_size,
                              hipStream_t stream) {
    const float*     pred   = (const float*)d_in[0];
    const long long* target = (const long long*)d_in[1];
    float*           out    = (float*)d_out;
    float*           ws     = (float*)d_ws;

    const int B = in_sizes[1];            // 65536
    const int C = in_sizes[0] / B;        // 1000

    const int nblocks = (B + 7) / 8;      // 8192 partials -> 32 KB of d_ws

    if (C == 1000) {
        hinge_rows_kernel<250><<<nblocks, 256, 0, stream>>>(pred, target, ws, B, C);
    } else {
        hinge_rows_kernel<0><<<nblocks, 256, 0, stream>>>(pred, target, ws, B, C);
    }
    reduce_mean_wmma<<<1, 32, 0, stream>>>(ws, out, nblocks, 1.0f / (float)B);
}